// DispCorrLayer_74603581931748
// MI455X (gfx1250) — compile-verified
//
#include <hip/hip_runtime.h>

// MI455X / gfx1250 disparity-correlation layer via V_WMMA_F32_16X16X4_F32.
// out[b,d,h,w] = (1/32) * sum_c in1[b,c,h,w] * in2[b,c,h,w+d-128]  (zero pad)
//
// Formulated as a banded Gram matrix per (b,h): G[w,u] = in1col(w).in2col(u)
// restricted to u in [w-128, w-1]; out[d,w] = G[w, w+d-128]/32.
// 16x16 G tiles = V_WMMA_F32_16X16X4_F32 accumulated over 8 k-steps (C=32).

typedef __attribute__((ext_vector_type(2))) float v2f;
typedef __attribute__((ext_vector_type(8))) float v8f;
typedef __attribute__((ext_vector_type(4))) float f4;

#define B_DIM   4
#define C_CH    32
#define H_DIM   256
#define W_DIM   512
#define D_DISP  128
#define CHUNK   128          // output w-chunk per workgroup
#define WIN     256          // in2 window: [w0-128, w0+127]
#define S_IN    34           // LDS stride (dwords) for transposed inputs (bank-conflict free)
#define S_OUT   132          // LDS stride (dwords) for output staging (16B aligned rows)

__global__ __launch_bounds__(256)
void disp_corr_wmma_kernel(const float* __restrict__ in1,
                           const float* __restrict__ in2,
                           float* __restrict__ out) {
    extern __shared__ float smem[];
    float* in1T = smem;                       // [CHUNK][S_IN]  w-major, channels contiguous
    float* in2T = smem + CHUNK * S_IN;        // [WIN][S_IN]
    float* outS = in2T + WIN * S_IN;          // [D_DISP][S_OUT]

    const int tid   = threadIdx.x;
    const int chunk = blockIdx.x;             // 0..3
    const int h     = blockIdx.y;             // 0..255
    const int b     = blockIdx.z;             // 0..3
    const int w0    = chunk * CHUNK;

    const size_t plane = (size_t)H_DIM * W_DIM;
    const float* p1 = in1 + (size_t)b * C_CH * plane + (size_t)h * W_DIM;
    const float* p2 = in2 + (size_t)b * C_CH * plane + (size_t)h * W_DIM;

    // ---- Stage in1 chunk transposed [w][c], pre-scaled by 1/C (coalesced reads) ----
    for (int e = tid; e < C_CH * CHUNK; e += 256) {
        int c  = e >> 7;          // e / 128
        int wl = e & 127;
        in1T[wl * S_IN + c] = p1[(size_t)c * plane + (w0 + wl)] * (1.0f / C_CH);
    }
    // ---- Stage in2 window transposed; branchless left zero padding ----
    // Clamp keeps the load unconditional (valid address); select zeroes pad lanes.
    for (int e = tid; e < C_CH * WIN; e += 256) {
        int c  = e >> 8;          // e / 256
        int ul = e & 255;
        int u  = w0 - D_DISP + ul;
        int uc = u < 0 ? 0 : u;
        float v = p2[(size_t)c * plane + uc];
        in2T[ul * S_IN + c] = (u < 0) ? 0.0f : v;
    }
    __syncthreads();

    // ---- Banded Gram computation via WMMA ----
    const int wave = tid >> 5;    // 0..7 : this wave's w-tile within the chunk
    const int lane = tid & 31;
    const int half = lane >> 4;   // 0/1
    const int n16  = lane & 15;
    const int wcol = wave * 16;   // base w column of this wave's tile

    // A fragments (16x4, M=w, K=c) for this wave's w-tile, all 8 k-steps.
    // Lane l, vgpr r holds A[M = l%16, K = 2*(l/16) + r] -> one ds_load_b64.
    v2f afrag[8];
    {
        const int wl = wcol + n16;
        #pragma unroll
        for (int s = 0; s < 8; ++s)
            afrag[s] = *(const v2f*)&in1T[wl * S_IN + 4 * s + 2 * half];
    }

    // One 16x16 G tile: 8 x V_WMMA_F32_16X16X4_F32 over the 32 channels.
    // B layout: lane l, vgpr r holds B[K = 2*(l/16)+r, N = l%16].
    auto compute_tile = [&](int jj) -> v8f {
        const int utb = (wave + jj) * 16;     // u-tile base in window coords
        v8f acc = {0.f, 0.f, 0.f, 0.f, 0.f, 0.f, 0.f, 0.f};
        #pragma unroll
        for (int s = 0; s < 8; ++s) {
            v2f bfrag = *(const v2f*)&in2T[(utb + n16) * S_IN + 4 * s + 2 * half];
            acc = __builtin_amdgcn_wmma_f32_16x16x4_f32(
                      false, afrag[s], false, bfrag,
                      (short)0, acc, false, false);
        }
        return acc;
    };

    // D layout: vgpr r, lane l -> (M = r + 8*half = w-in-tile, N = n16 = u-in-tile).
    // outS dword index = d*S_OUT + wcol + m, d = 16*jj + n16 - m, m = r + 8*half
    //                  = [S_OUT*n16 + wcol - 1048*half] + 2112*jj - 131*r.
    // The bracket is per-lane invariant -> base pointer; the rest is a literal.
    float* q = outS + (S_OUT * n16 + wcol - 1048 * half);

    // Edge tiles jj=0 (keep d = n16-m >= 0) and jj=8 (keep d = 128+n16-m < 128)
    // have complementary masks: every (r,lane) writes exactly one of the two.
    // Merge them into unconditional stores with two selects -> no EXEC churn.
    {
        v8f acc0 = compute_tile(0);
        v8f acc8 = compute_tile(8);
        #pragma unroll
        for (int r = 0; r < 8; ++r) {
            bool lead = (n16 >= r + 8 * half);
            float v   = lead ? acc0[r] : acc8[r];
            int  idx  = (lead ? 0 : 2112 * 8) - 131 * r;
            q[idx] = v;
        }
    }
    #pragma unroll
    for (int jj = 1; jj <= 7; ++jj) {         // interior tiles: d in [1,127] always
        v8f acc = compute_tile(jj);
        #pragma unroll
        for (int r = 0; r < 8; ++r)
            q[2112 * jj - 131 * r] = acc[r];
    }
    __syncthreads();

    // ---- Coalesced float4 writeback of the 128x128 chunk ----
    float* po = out + (size_t)b * D_DISP * plane + (size_t)h * W_DIM + w0;
    for (int i = tid; i < D_DISP * (CHUNK / 4); i += 256) {
        int d  = i >> 5;                      // row (disparity)
        int wv = (i & 31) * 4;                // vec4 column (S_OUT%4==0 -> 16B aligned)
        f4 v = *(const f4*)&outS[d * S_OUT + wv];
        *(f4*)&po[(size_t)d * plane + wv] = v;
    }
}

extern "C" void kernel_launch(void* const* d_in, const int* in_sizes, int n_in,
                              void* d_out, int out_size, void* d_ws, size_t ws_size,
                              hipStream_t stream) {
    (void)in_sizes; (void)n_in; (void)out_size; (void)d_ws; (void)ws_size;
    const float* in1 = (const float*)d_in[0];
    const float* in2 = (const float*)d_in[1];
    float* out = (float*)d_out;

    const size_t shmem =
        (size_t)(CHUNK * S_IN + WIN * S_IN + D_DISP * S_OUT) * sizeof(float); // 119808 B

    static bool attr_set = false;
    if (!attr_set) {
        hipFuncSetAttribute((const void*)disp_corr_wmma_kernel,
                            hipFuncAttributeMaxDynamicSharedMemorySize,
                            (int)shmem);
        attr_set = true;
    }

    dim3 grid(W_DIM / CHUNK, H_DIM, B_DIM);   // (4, 256, 4) = 4096 blocks
    disp_corr_wmma_kernel<<<grid, 256, shmem, stream>>>(in1, in2, out);
}